// RealSpaceEGNNEncoder_31714038514065
// MI455X (gfx1250) — compile-verified
//
// CDNA5 / gfx1250 e3nn-style equivariant GNN encoder.
// fp32 end-to-end. All tensor-product / linear contractions -> V_WMMA_F32_16X16X4_F32
// (M=16 nodes|edges, K streamed by 4, N=m3). Weights are streamed into LDS with the
// Tensor Data Mover (tensor_load_to_lds + s_wait_tensorcnt) and reused across all
// output m-components; WMMA consumes A and B operands from LDS.
// Workspace (floats): h[2.4M] | agg[2.4M] | pooled[4096] | cnt[64]  (~19.3MB)

#include <hip/hip_runtime.h>
#include <cmath>
#include <complex>

#define NNODES 10000
#define NEDGES 160000
#define NGRAPHS 64
#define HDIM 240
#define NLAYERS 6
#define MSG_WSZ 10496
#define UPD_WSZ 561152
#define LIN_WSZ 5376
#define HST 244   // padded LDS row stride for 240-wide tiles
#define GST 68    // padded LDS row stride for GEMM A-operand tiles (K<=64)

typedef __attribute__((ext_vector_type(2))) float v2f;
typedef __attribute__((ext_vector_type(8))) float v8f;
typedef __attribute__((ext_vector_type(4))) unsigned u32x4;
typedef __attribute__((ext_vector_type(4))) int i32x4;
typedef __attribute__((ext_vector_type(8))) int i32x8;

#if defined(__has_builtin)
# if __has_builtin(__builtin_amdgcn_tensor_load_to_lds) && __has_builtin(__builtin_amdgcn_s_wait_tensorcnt)
#  define USE_TDM 1
# endif
#endif
#ifndef USE_TDM
# define USE_TDM 0
#endif

// Real Wigner-3j tables, computed host-side, passed by value (kernarg).
// layout: c000@0(1) c011@1(9) c101@10(9) c110@19(9) c112@28(45) c022@73(25)
//         c202@98(25) c121@123(45) c211@168(45) c220@213(25) c222@238(125) => 363
struct CGPack { float c[363]; };

struct Path { int off1,l1,m1,lg1, off2,l2,m2,lg2, off3,l3,m3, wofs, cgofs; float alpha; };

// ---- message paths (EDGE multiplicity is 1 -> W is [m1 x m3]) ----
__constant__ Path MSGP[7] = {
  {  0,0,64,6,  0,0,1,0,   0,0,64,    0,   0, 0.08838834764831845f}, // (0,0,0)
  {  0,0,64,6,  0,1,1,0,  64,1,32, 4096,   1, 0.07216878364870323f}, // (0,1,1)
  { 64,1,32,5,  0,0,1,0,  64,1,32, 6144,  10, 0.10206207261596575f}, // (1,0,1)
  { 64,1,32,5,  0,1,1,0,   0,0,64, 7168,  19, 0.125f              }, // (1,1,0)
  { 64,1,32,5,  0,1,1,0, 160,2,16, 9216,  28, 0.125f              }, // (1,1,2)
  {160,2,16,4,  0,0,1,0, 160,2,16, 9728,  98, 0.17677669529663687f}, // (2,0,2)
  {160,2,16,4,  0,1,1,0,  64,1,32, 9984, 168, 0.14433756729740643f}, // (2,1,1)
};

// ---------------- TDM weight staging (global -> LDS, 1-D tile) ----------------
__device__ __forceinline__ void twait(){
#if USE_TDM
  __builtin_amdgcn_s_wait_tensorcnt(0);
#endif
}
__device__ __forceinline__ void tdm_stage(float* wbuf, unsigned lds_off,
                                          const float* __restrict__ src, int nfloats){
#if USE_TDM
  unsigned long long g = (unsigned long long)(size_t)src;
  u32x4 g0;
  g0.x = 1u;                                    // count=1, user descriptor
  g0.y = lds_off;                               // LDS byte address of tile start
  g0.z = (unsigned)(g & 0xffffffffull);         // global_addr[31:0]
  g0.w = (unsigned)((g >> 32) & 0x01ffffffull) | 0x80000000u; // addr[56:32] | type=2
  i32x8 g1;
  g1[0] = 0x20000;                              // data_size=4B, mask=0, no pad/iter
  g1[1] = (int)(((unsigned)nfloats & 0xffffu) << 16);          // tensor_dim0 lo16
  g1[2] = (int)(((unsigned)nfloats >> 16) | (1u << 16));       // tensor_dim0 hi | tensor_dim1=1
  g1[3] = (int)((unsigned)nfloats << 16);       // tile_dim0 (tensor_dim1 hi=0)
  g1[4] = 0;                                    // tile_dim1=0, tile_dim2=0 -> 1-D tile
  g1[5] = nfloats;                              // tensor_dim0_stride lo
  g1[6] = 0; g1[7] = 0;
  i32x4 z4 = {0,0,0,0};
# if defined(__clang_major__) && __clang_major__ >= 23
  i32x8 z8 = {0,0,0,0,0,0,0,0};
  __builtin_amdgcn_tensor_load_to_lds(g0, g1, z4, z4, z8, 0);
# else
  __builtin_amdgcn_tensor_load_to_lds(g0, g1, z4, z4, 0);
# endif
#else
  (void)lds_off;
  for (int t = threadIdx.x; t < nfloats; t += 32) wbuf[t] = src[t];
#endif
}

// ---------------- WMMA helpers (f32 16x16x4) ----------------
// A (16x4, MxK): VGPR0={K0 lanes0-15, K2 lanes16-31}, VGPR1={K1,K3}
__device__ __forceinline__ v2f ldsA(const float* p, int stride, int k0, int lane){
  int m = lane & 15, kh = (lane >> 4) << 1;
  v2f a; a.x = p[m*stride + k0 + kh]; a.y = p[m*stride + k0 + kh + 1]; return a;
}
// B (4x16, KxN), row-major [K][N] in LDS with given stride
__device__ __forceinline__ v2f ldsB(const float* p, int stride, int k0, int n0, int lane){
  int n = lane & 15, kh = (lane >> 4) << 1;
  v2f b; b.x = p[(k0+kh)*stride + n0 + n]; b.y = p[(k0+kh+1)*stride + n0 + n]; return b;
}
__device__ __forceinline__ v8f wmma4(v2f a, v2f b, v8f c){
  return __builtin_amdgcn_wmma_f32_16x16x4_f32(false, a, false, b, (short)0, c, false, false);
}

// ---------------- irrep-linear (templated, TDM-staged, K-chunked) ----------------
// dst[e][OFF+w*NL+k] += ALPHA * sum_u src[e][OFF+u*NL+k]*W[u][w]
template<int OFF,int L,int M,int CAP>
__device__ __forceinline__ void lin_irrep(const float* __restrict__ W, float ALPHA,
    const float* src, float* dst, float* gbuf, float* wbuf, unsigned wlds, int lane)
{
  constexpr int NL = 2*L+1, NTN = M/16;
  constexpr int CH = (M*M > CAP) ? (CAP/M) : M;
  constexpr int NCH = M/CH;
  v8f accs[NL][NTN];
  #pragma unroll
  for (int k=0;k<NL;++k)
    #pragma unroll
    for (int nt=0;nt<NTN;++nt) accs[k][nt] = {};
  for (int c=0;c<NCH;++c){
    __syncthreads();                                   // retire prior wbuf/gbuf reads
    tdm_stage(wbuf, wlds, W + c*CH*M, CH*M);
    #pragma unroll
    for (int k=0;k<NL;++k){
      __syncthreads();
      for (int t=lane; t<16*CH; t+=32){
        int e = t / CH, ul = t % CH, u = c*CH + ul;
        gbuf[e*GST + c*CH + ul] = src[e*HST + OFF + u*NL + k];
      }
      __syncthreads();
      twait();
      #pragma unroll
      for (int nt=0;nt<NTN;++nt)
        for (int ks=0;ks<CH/4;++ks)
          accs[k][nt] = wmma4(ldsA(gbuf,GST,c*CH+ks*4,lane),
                              ldsB(wbuf,M,ks*4,nt*16,lane), accs[k][nt]);
    }
  }
  int nn = lane & 15, mb = (lane >> 4) * 8;
  #pragma unroll
  for (int k=0;k<NL;++k)
    #pragma unroll
    for (int nt=0;nt<NTN;++nt){
      int w = nt*16 + nn;
      #pragma unroll
      for (int j=0;j<8;++j)
        dst[(mb+j)*HST + OFF + w*NL + k] += ALPHA*accs[k][nt][j];
    }
}

// ---------------- update TP path (templated, TDM-staged, u-outer) ----------------
template<int OFF1,int L1,int M1,int OFF2,int L2,int M2,int OFF3,int L3,int M3,int CAP>
__device__ __forceinline__ void upd_path(const float* __restrict__ W,
    const float* cgs, int cgofs, float alpha,
    const float* at, const float* bt, float* ut,
    float* zb, float* ca, float* wbuf, unsigned wlds, int lane)
{
  constexpr int N1=2*L1+1, N2=2*L2+1, N3=2*L3+1, NTN=M3/16;
  constexpr int CH  = (M2*M3 > CAP) ? (CAP/M3) : M2;   // K-chunk (v rows per stage)
  constexpr int NCH = M2/CH;
  v8f accs[N3][NTN];
  #pragma unroll
  for (int k=0;k<N3;++k)
    #pragma unroll
    for (int nt=0;nt<NTN;++nt) accs[k][nt] = {};
  for (int u=0; u<M1; ++u){
    const float* Wu = W + u*M2*M3;
    #pragma unroll
    for (int c=0;c<NCH;++c){
      __syncthreads();                                 // retire prior wbuf/zb reads
      tdm_stage(wbuf, wlds, Wu + c*CH*M3, CH*M3);
      #pragma unroll
      for (int k3=0;k3<N3;++k3){
        __syncthreads();
        if (lane < 16){                                // ca[n][j] = sum_i a[n,u,i]*C[i,j,k3]
          #pragma unroll
          for (int j=0;j<N2;++j){
            float s=0.f;
            #pragma unroll
            for (int i=0;i<N1;++i)
              s += at[lane*HST + OFF1 + u*N1 + i] * cgs[cgofs + (i*N2+j)*N3 + k3];
            ca[lane*8+j]=s;
          }
        }
        __syncthreads();
        for (int t=lane; t<16*CH; t+=32){              // z[n][v] = sum_j ca[n][j]*b[n,v,j]
          int n = t / CH, vl = t % CH, v = c*CH + vl;
          float s=0.f;
          #pragma unroll
          for (int j=0;j<N2;++j) s += ca[n*8+j]*bt[n*HST + OFF2 + v*N2 + j];
          zb[n*GST + v] = s;
        }
        __syncthreads();
        twait();
        #pragma unroll
        for (int nt=0;nt<NTN;++nt)
          for (int ks=0;ks<CH/4;++ks)
            accs[k3][nt] = wmma4(ldsA(zb,GST,c*CH+ks*4,lane),
                                 ldsB(wbuf,M3,ks*4,nt*16,lane), accs[k3][nt]);
      }
    }
  }
  int nn = lane & 15, mb = (lane >> 4) * 8;
  #pragma unroll
  for (int k3=0;k3<N3;++k3)
    #pragma unroll
    for (int nt=0;nt<NTN;++nt){
      int w = nt*16 + nn;
      #pragma unroll
      for (int j=0;j<8;++j)
        ut[(mb+j)*HST + OFF3 + w*N3 + k3] += alpha*accs[k3][nt][j];
    }
}

// ---------------- kernels ----------------
__global__ void init_h(const float* __restrict__ x, const float* __restrict__ wi,
                       float* __restrict__ h){
  int t = blockIdx.x*blockDim.x + threadIdx.x;
  if (t >= NNODES*HDIM) return;
  int n = t / HDIM, c = t - n*HDIM;
  float v = 0.f;
  if (c < 64){
    float s = 0.f;
    for (int d=0; d<3; ++d) s += x[n*3+d]*wi[d*64+c];
    v = s * 0.57735026918962576f; // 1/sqrt(3)
  }
  h[t] = v;
}

__global__ void zero_f(float* __restrict__ p, int n){
  int t = blockIdx.x*blockDim.x + threadIdx.x;
  if (t < n) p[t] = 0.f;
}

// one wave per 16-edge tile: TP(h[col], edge_attr) + linear(h[col]) -> atomic scatter to agg[row]
__global__ __launch_bounds__(32) void msg_kernel(const float* __restrict__ h,
    float* __restrict__ agg, const float* __restrict__ pos, const int* __restrict__ ei,
    const float* __restrict__ wtp, const float* __restrict__ wlin, CGPack cg)
{
  __shared__ float hjt[16*HST];
  __shared__ float msgt[16*HST];
  __shared__ float gbuf[16*GST];
  __shared__ float wbuf[4096];
  __shared__ float cb[16*8];
  __shared__ float eb[16*4];
  __shared__ float cgs[363];
  __shared__ int ridx[16], cidx[16];
  int lane = threadIdx.x;
  int e0 = blockIdx.x * 16;
  unsigned wlds = (unsigned)(size_t)(void*)wbuf;
  for (int t=lane;t<363;t+=32) cgs[t]=cg.c[t];
  if (lane < 16){ ridx[lane]=ei[e0+lane]; cidx[lane]=ei[NEDGES+e0+lane]; }
  __syncthreads();
  if (lane < 16){
    int r=ridx[lane], c=cidx[lane];
    float rx=pos[r*3+0]-pos[c*3+0], ry=pos[r*3+1]-pos[c*3+1], rz=pos[r*3+2]-pos[c*3+2];
    float d = sqrtf(rx*rx+ry*ry+rz*rz);
    float inv = 1.0f / fmaxf(d, 1e-8f);
    eb[lane*4+0]=d; eb[lane*4+1]=rx*inv; eb[lane*4+2]=ry*inv; eb[lane*4+3]=rz*inv;
  }
  for (int e=0;e<16;++e){
    int c=cidx[e];
    for (int t=lane;t<HDIM;t+=32){ hjt[e*HST+t]=h[c*HDIM+t]; msgt[e*HST+t]=0.f; }
  }
  __syncthreads();

  for (int p=0;p<7;++p){
    Path P = MSGP[p];
    int n1=2*P.l1+1, n2=2*P.l2+1, n3=2*P.l3+1;
    int ebase = (P.l2==0)?0:1;
    __syncthreads();                                   // retire prior wbuf reads
    tdm_stage(wbuf, wlds, wtp + P.wofs, P.m1*P.m3);    // stage W once per path
    for (int k3=0;k3<n3;++k3){
      __syncthreads();
      if (lane<16){
        for (int i=0;i<n1;++i){
          float s=0.f;
          for (int j=0;j<n2;++j) s += cgs[P.cgofs + (i*n2+j)*n3 + k3] * eb[lane*4+ebase+j];
          cb[lane*8+i]=s;
        }
      }
      __syncthreads();
      for (int t=lane; t<16*P.m1; t+=32){              // g[e][u] = sum_i a[e,u,i]*cb[e,i]
        int e = t >> P.lg1, u = t & (P.m1-1);
        float s=0.f;
        for (int i=0;i<n1;++i) s += hjt[e*HST + P.off1 + u*n1 + i] * cb[e*8+i];
        gbuf[e*GST+u]=s;
      }
      __syncthreads();
      twait();
      for (int nt=0; nt<(P.m3>>4); ++nt){
        v8f acc = {};
        for (int ks=0; ks<(P.m1>>2); ++ks)
          acc = wmma4(ldsA(gbuf,GST,ks*4,lane), ldsB(wbuf,P.m3,ks*4,nt*16,lane), acc);
        int w = nt*16 + (lane&15); int mb=(lane>>4)*8;
        #pragma unroll
        for (int j=0;j<8;++j)
          msgt[(mb+j)*HST + P.off3 + w*n3 + k3] += P.alpha * acc[j];
      }
    }
  }
  lin_irrep<  0,0,64,4096>(wlin+0,    0.125f,               hjt, msgt, gbuf, wbuf, wlds, lane);
  lin_irrep< 64,1,32,4096>(wlin+4096, 0.17677669529663687f, hjt, msgt, gbuf, wbuf, wlds, lane);
  lin_irrep<160,2,16,4096>(wlin+5120, 0.25f,                hjt, msgt, gbuf, wbuf, wlds, lane);
  __syncthreads();
  for (int e=0;e<16;++e){
    int r = ridx[e];
    for (int t=lane;t<HDIM;t+=32) atomicAdd(&agg[r*HDIM+t], msgt[e*HST+t]);
  }
}

// one wave per 16-node tile: h += TP(h, agg) + linear(h)
__global__ __launch_bounds__(32) void upd_kernel(float* __restrict__ h,
    const float* __restrict__ agg, const float* __restrict__ wtp,
    const float* __restrict__ wlin, CGPack cg)
{
  __shared__ float at[16*HST], bt[16*HST], ut[16*HST];
  __shared__ float zb[16*GST];
  __shared__ float wbuf[2048];
  __shared__ float ca[16*8];
  __shared__ float cgs[363];
  int lane = threadIdx.x;
  int nb = blockIdx.x * 16;
  unsigned wlds = (unsigned)(size_t)(void*)wbuf;
  for (int t=lane;t<363;t+=32) cgs[t]=cg.c[t];
  for (int n=0;n<16;++n)
    for (int t=lane;t<HDIM;t+=32){
      at[n*HST+t]=h[(nb+n)*HDIM+t];
      bt[n*HST+t]=agg[(nb+n)*HDIM+t];
      ut[n*HST+t]=0.f;
    }
  __syncthreads();

  upd_path<  0,0,64,   0,0,64,   0,0,64, 2048>(wtp+0,      cgs,   0, 0.00902109795608790f, at,bt,ut,zb,ca,wbuf,wlds,lane);
  upd_path<  0,0,64,  64,1,32,  64,1,32, 2048>(wtp+262144, cgs,   1, 0.01104854345603981f, at,bt,ut,zb,ca,wbuf,wlds,lane);
  upd_path<  0,0,64, 160,2,16, 160,2,16, 2048>(wtp+327680, cgs,  73, 0.015625f,            at,bt,ut,zb,ca,wbuf,wlds,lane);
  upd_path< 64,1,32,   0,0,64,  64,1,32, 2048>(wtp+344064, cgs,  10, 0.01104854345603981f, at,bt,ut,zb,ca,wbuf,wlds,lane);
  upd_path< 64,1,32,  64,1,32,   0,0,64, 2048>(wtp+409600, cgs,  19, 0.01804219591217581f, at,bt,ut,zb,ca,wbuf,wlds,lane);
  upd_path< 64,1,32,  64,1,32, 160,2,16, 2048>(wtp+475136, cgs,  28, 0.015625f,            at,bt,ut,zb,ca,wbuf,wlds,lane);
  upd_path< 64,1,32, 160,2,16,  64,1,32, 2048>(wtp+491520, cgs, 123, 0.02209708691207961f, at,bt,ut,zb,ca,wbuf,wlds,lane);
  upd_path<160,2,16,   0,0,64, 160,2,16, 2048>(wtp+507904, cgs,  98, 0.015625f,            at,bt,ut,zb,ca,wbuf,wlds,lane);
  upd_path<160,2,16,  64,1,32,  64,1,32, 2048>(wtp+524288, cgs, 168, 0.02209708691207961f, at,bt,ut,zb,ca,wbuf,wlds,lane);
  upd_path<160,2,16, 160,2,16,   0,0,64, 2048>(wtp+540672, cgs, 213, 0.03608439182435162f, at,bt,ut,zb,ca,wbuf,wlds,lane);
  upd_path<160,2,16, 160,2,16, 160,2,16, 2048>(wtp+557056, cgs, 238, 0.03125f,             at,bt,ut,zb,ca,wbuf,wlds,lane);

  lin_irrep<  0,0,64,2048>(wlin+0,    0.125f,               at, ut, zb, wbuf, wlds, lane);
  lin_irrep< 64,1,32,2048>(wlin+4096, 0.17677669529663687f, at, ut, zb, wbuf, wlds, lane);
  lin_irrep<160,2,16,2048>(wlin+5120, 0.25f,                at, ut, zb, wbuf, wlds, lane);

  __syncthreads();
  for (int n=0;n<16;++n)
    for (int t=lane;t<HDIM;t+=32)
      h[(nb+n)*HDIM+t] = at[n*HST+t] + ut[n*HST+t];
}

__global__ void pool_kernel(const float* __restrict__ h, const int* __restrict__ batch,
                            float* __restrict__ pooled, float* __restrict__ cnt){
  int t = blockIdx.x*blockDim.x + threadIdx.x;
  if (t >= NNODES*64) return;
  int n = t >> 6, c = t & 63;
  int g = batch[n];
  atomicAdd(&pooled[g*64+c], h[n*HDIM+c]);
  if (c==0) atomicAdd(&cnt[g], 1.0f);
}

__global__ void final_kernel(const float* __restrict__ pooled, const float* __restrict__ cnt,
                             const float* __restrict__ wf, float* __restrict__ out){
  int t = blockIdx.x*blockDim.x + threadIdx.x;
  if (t >= NGRAPHS*128) return;
  int g = t >> 7, o = t & 127;
  float inv = 1.0f / fmaxf(cnt[g], 1.0f);
  float s = 0.f;
  for (int c=0;c<64;++c) s += pooled[g*64+c]*wf[c*128+o];
  out[t] = s*inv*0.125f;  // /sqrt(64)
}

// ---------------- host: real Wigner-3j tables (same algorithm as reference) ----------------
typedef std::complex<double> cpx;
static double hfact(int n){ double r=1.0; for(int i=2;i<=n;++i) r*=(double)i; return r; }
static double hcg(int j1,int m1,int j2,int m2,int j3,int m3){
  if (m1+m2!=m3) return 0.0;
  double pre = std::sqrt((2.0*j3+1.0)*hfact(j3+j1-j2)*hfact(j3-j1+j2)*hfact(j1+j2-j3)/hfact(j1+j2+j3+1));
  pre *= std::sqrt(hfact(j3+m3)*hfact(j3-m3)*hfact(j1-m1)*hfact(j1+m1)*hfact(j2-m2)*hfact(j2+m2));
  double s=0.0;
  for (int k=0;k<=j1+j2-j3;++k){
    int a0=k, a1=j1+j2-j3-k, a2=j1-m1-k, a3=j2+m2-k, a4=j3-j2+m1+k, a5=j3-j1-m2+k;
    if (a0<0||a1<0||a2<0||a3<0||a4<0||a5<0) continue;
    double d = hfact(a0)*hfact(a1)*hfact(a2)*hfact(a3)*hfact(a4)*hfact(a5);
    s += ((k&1)? -1.0:1.0)/d;
  }
  return pre*s;
}
static void makeA(int l, cpx A[5][5]){
  for (int a=0;a<5;++a) for (int b=0;b<5;++b) A[a][b]=cpx(0,0);
  A[l][l]=cpx(1,0);
  double r2 = 1.0/std::sqrt(2.0);
  for (int m=1;m<=l;++m){
    double s = (m&1)? -1.0:1.0;
    A[l+m][l+m]=cpx(s*r2,0);
    A[l+m][l-m]=cpx(r2,0);
    A[l-m][l+m]=cpx(0,-s*r2);
    A[l-m][l-m]=cpx(0,r2);
  }
}
static void real_w3j_host(int l1,int l2,int l3,float* out){
  int n1=2*l1+1,n2=2*l2+1,n3=2*l3+1;
  cpx Wc[5][5][5];
  for (int a=0;a<5;++a)for(int b=0;b<5;++b)for(int c=0;c<5;++c) Wc[a][b][c]=cpx(0,0);
  for (int m1=-l1;m1<=l1;++m1) for (int m2=-l2;m2<=l2;++m2){
    int m3=-(m1+m2);
    if (m3<-l3 || m3>l3) continue;
    int e = l1-l2-m3;
    double sgn = (((e%2)+2)%2)? -1.0:1.0;
    Wc[m1+l1][m2+l2][m3+l3] = cpx(sgn/std::sqrt(2.0*l3+1.0)*hcg(l1,m1,l2,m2,l3,-m3),0);
  }
  cpx A1[5][5],A2[5][5],A3[5][5];
  makeA(l1,A1); makeA(l2,A2); makeA(l3,A3);
  double T[5][5][5]; double nrm=0.0;
  for (int a=0;a<n1;++a)for(int b=0;b<n2;++b)for(int c=0;c<n3;++c){
    cpx s(0,0);
    for (int d=0;d<n1;++d)for(int e2=0;e2<n2;++e2)for(int f=0;f<n3;++f)
      s += A1[a][d]*A2[b][e2]*A3[c][f]*Wc[d][e2][f];
    T[a][b][c]=s.real();
    nrm += T[a][b][c]*T[a][b][c];
  }
  nrm = std::sqrt(nrm);
  for (int a=0;a<n1;++a)for(int b=0;b<n2;++b)for(int c=0;c<n3;++c)
    out[(a*n2+b)*n3+c] = (float)(T[a][b][c]/nrm);
}
static void build_cg(CGPack& P){
  real_w3j_host(0,0,0,P.c+0);
  real_w3j_host(0,1,1,P.c+1);
  real_w3j_host(1,0,1,P.c+10);
  real_w3j_host(1,1,0,P.c+19);
  real_w3j_host(1,1,2,P.c+28);
  real_w3j_host(0,2,2,P.c+73);
  real_w3j_host(2,0,2,P.c+98);
  real_w3j_host(1,2,1,P.c+123);
  real_w3j_host(2,1,1,P.c+168);
  real_w3j_host(2,2,0,P.c+213);
  real_w3j_host(2,2,2,P.c+238);
}

extern "C" void kernel_launch(void* const* d_in, const int* in_sizes, int n_in,
                              void* d_out, int out_size, void* d_ws, size_t ws_size,
                              hipStream_t stream)
{
  (void)in_sizes; (void)n_in; (void)out_size; (void)ws_size;
  const float* x         = (const float*)d_in[0];
  const float* pos       = (const float*)d_in[1];
  const float* w_init    = (const float*)d_in[2];
  const float* w_tp_msg  = (const float*)d_in[3];
  const float* w_lin_msg = (const float*)d_in[4];
  const float* w_tp_upd  = (const float*)d_in[5];
  const float* w_lin_upd = (const float*)d_in[6];
  const float* w_final   = (const float*)d_in[7];
  const int*   ei        = (const int*)d_in[8];
  const int*   batch     = (const int*)d_in[9];
  float* out = (float*)d_out;
  float* ws  = (float*)d_ws;
  float* h      = ws;                      // 2,400,000 floats
  float* agg    = ws + 2400000;            // 2,400,000 floats
  float* pooled = ws + 4800000;            // 4,096 floats
  float* cnt    = pooled + 64*64;          // 64 floats

  CGPack cg;
  build_cg(cg);

  init_h<<<(NNODES*HDIM+255)/256, 256, 0, stream>>>(x, w_init, h);
  for (int L=0; L<NLAYERS; ++L){
    zero_f<<<(NNODES*HDIM+255)/256, 256, 0, stream>>>(agg, NNODES*HDIM);
    msg_kernel<<<NEDGES/16, 32, 0, stream>>>(h, agg, pos, ei,
        w_tp_msg + (size_t)L*MSG_WSZ, w_lin_msg + (size_t)L*LIN_WSZ, cg);
    upd_kernel<<<NNODES/16, 32, 0, stream>>>(h, agg,
        w_tp_upd + (size_t)L*UPD_WSZ, w_lin_upd + (size_t)L*LIN_WSZ, cg);
  }
  zero_f<<<(64*64+64+255)/256, 256, 0, stream>>>(pooled, 64*64+64);
  pool_kernel<<<(NNODES*64+255)/256, 256, 0, stream>>>(h, batch, pooled, cnt);
  final_kernel<<<(NGRAPHS*128+255)/256, 256, 0, stream>>>(pooled, cnt, w_final, out);
}